// HGTLayer_42374147342881
// MI455X (gfx1250) — compile-verified
//
#include <hip/hip_runtime.h>
#include <math.h>

typedef __attribute__((ext_vector_type(2))) float v2f;
typedef __attribute__((ext_vector_type(8))) float v8f;
typedef unsigned int v4u __attribute__((ext_vector_type(4)));
typedef int v8i __attribute__((ext_vector_type(8)));
typedef int v4i __attribute__((ext_vector_type(4)));

#define N_NODES 100000
#define E_EDGES 400000
#define PLD 1280          // row stride of fused projection buffer P = [q|kr0|kr1|vr0|vr1]
#define MB 48             // GEMM M-tile per workgroup (48*260*4 = 49,920 B LDS)
#define ALD 260           // padded LDS row stride (floats): stride%64==4 -> conflict-free

// ---------------------------------------------------------------------------
// Build fused projection weights:
//   Wbig[i, 0:256]            = Wq[i, :]
//   Wbig[i, 256+r*256+h*32+e] = sum_d Wk[i,h*32+d] * rel_att[r,h,d,e]
//   Wbig[i, 768+r*256+h*32+e] = sum_d Wv[i,h*32+d] * rel_msg[r,h,d,e]
// Row 256 computes the matching fused biases from bq/bk/bv.
// ---------------------------------------------------------------------------
__global__ __launch_bounds__(256) void build_wbig(
    const float* __restrict__ Wk, const float* __restrict__ bk,
    const float* __restrict__ Wq, const float* __restrict__ bq,
    const float* __restrict__ Wv, const float* __restrict__ bv,
    const float* __restrict__ rel_att, const float* __restrict__ rel_msg,
    float* __restrict__ Wbig, float* __restrict__ BiasBig) {
  int idx = blockIdx.x * blockDim.x + threadIdx.x;
  if (idx >= 257 * 1280) return;
  int row = idx / 1280;          // 256 == bias row
  int col = idx - row * 1280;
  bool is_bias = (row == 256);
  float out;
  if (col < 256) {
    out = is_bias ? bq[col] : Wq[row * 256 + col];
  } else {
    int sec = (col - 256) >> 8;  // 0: att r0, 1: att r1, 2: msg r0, 3: msg r1
    int c = (col - 256) & 255;
    int hd = c >> 5, e = c & 31;
    int r = sec & 1;
    const float* rel = (sec < 2) ? rel_att : rel_msg;
    const float* W   = (sec < 2) ? Wk : Wv;
    const float* bb  = (sec < 2) ? bk : bv;
    float s = 0.f;
    #pragma unroll 8
    for (int d = 0; d < 32; ++d) {
      float w = is_bias ? bb[hd * 32 + d] : W[row * 256 + hd * 32 + d];
      s += w * rel[(((r * 8 + hd) * 32 + d) * 32) + e];
    }
    out = s;
  }
  if (is_bias) BiasBig[col] = out;
  else         Wbig[row * 1280 + col] = out;
}

__global__ void zero_f32(float* __restrict__ p, size_t n) {
  size_t i = (size_t)blockIdx.x * blockDim.x + threadIdx.x;
  size_t stride = (size_t)gridDim.x * blockDim.x;
  for (; i < n; i += stride) p[i] = 0.f;
}

// ---------------------------------------------------------------------------
// FP32 WMMA GEMM, K fixed at 256:  C = scale*(A@B) + bias + (res ? res : 0)
// Workgroup: 256 threads = 8 waves. A-tile (MB x 256) staged into LDS by the
// Tensor Data Mover: one tensor_load_to_lds with hardware row padding
// (pad_interval=256 DW, pad_amount=4 DW -> LDS stride 260, bank-conflict-free).
// Each wave: one 16-wide column strip, 3 stacked 16x16 M-tiles,
// 64 k-steps of v_wmma_f32_16x16x4_f32 with B-fragment reuse across M-tiles.
// ---------------------------------------------------------------------------
__global__ __launch_bounds__(256) void wmma_gemm_k256(
    const float* __restrict__ A, int lda,
    const float* __restrict__ B, int ldb,
    const float* __restrict__ bias,
    const float* __restrict__ res, int ldres,
    float scale,
    float* __restrict__ C, int ldc, int M) {
  __shared__ float As[MB * ALD];
  const int m_base = blockIdx.x * MB;
  const int n0 = blockIdx.y * 128 + (threadIdx.x >> 5) * 16;

#if defined(__has_builtin) && __has_builtin(__builtin_amdgcn_tensor_load_to_lds) && \
    __has_builtin(__builtin_amdgcn_s_wait_tensorcnt)
  // ---- TDM path: single-instruction 2D tile DMA (wave 0 issues it) ----
  if (threadIdx.x < 32) {
    unsigned long long gaddr = (unsigned long long)(uintptr_t)(A + (size_t)m_base * lda);
    unsigned int ldsa = (unsigned int)(uintptr_t)&As[0];   // low 32 bits = LDS offset
    int rows_rem = M - m_base;                             // OOB rows read back as 0
    v4u g0;
    g0[0] = 1u;                                            // count=1 (valid descriptor)
    g0[1] = ldsa;                                          // lds_addr
    g0[2] = (unsigned int)gaddr;                           // global_addr[31:0]
    g0[3] = (unsigned int)((gaddr >> 32) & 0x1FFFFFFu)     // global_addr[56:32]
          | (2u << 30);                                    // type=2 ("image")
    v8i g1;
    g1[0] = (2 << 16)            // data_size = 4 bytes
          | (1 << 20)            // pad_enable
          | (7 << 22)            // pad_interval: 256 DWORDs
          | (3 << 25);           // pad_amount: 4 DWORDs  -> row stride 260 floats
    g1[1] = (lda & 0xFFFF) << 16;                          // tensor_dim0[15:0]
    g1[2] = ((lda >> 16) & 0xFFFF) | ((rows_rem & 0xFFFF) << 16);  // tdim0 hi | tdim1 lo
    g1[3] = ((rows_rem >> 16) & 0xFFFF) | (256 << 16);     // tdim1 hi | tile_dim0=256
    g1[4] = MB;                                            // tile_dim1=48, tile_dim2=0
    g1[5] = lda;                                           // tensor_dim0_stride[31:0]
    g1[6] = 0;
    g1[7] = 0;
    v4i zero4 = {0, 0, 0, 0};
    v8i zero8 = {0, 0, 0, 0, 0, 0, 0, 0};
    __builtin_amdgcn_tensor_load_to_lds(g0, g1, zero4, zero4, zero8, 0);
    __builtin_amdgcn_s_wait_tensorcnt(0);
  }
#else
  // ---- fallback: cooperative, coalesced float4 load of A tile ----
  for (int fi = threadIdx.x; fi < MB * 64; fi += 256) {
    int row = fi >> 6;
    int col = (fi & 63) * 4;
    int gr = m_base + row;
    if (gr > M - 1) gr = M - 1;
    float4 v = *(const float4*)&A[(size_t)gr * lda + col];
    *(float4*)&As[row * ALD + col] = v;
  }
#endif
  __syncthreads();

  const int lane = threadIdx.x & 31;
  const int l  = lane & 15;   // M-row / N-col within fragment
  const int hf = lane >> 4;   // half: selects K pair (A) / M+8 (D)

  v8f acc0 = {}, acc1 = {}, acc2 = {};
  const float* Bp = B + (size_t)(n0 + l);
  #pragma unroll 4
  for (int k0 = 0; k0 < 256; k0 += 4) {
    const int kk = k0 + 2 * hf;
    v2f b;
    b.x = Bp[(size_t)kk * ldb];
    b.y = Bp[(size_t)(kk + 1) * ldb];
    v2f a0 = *(const v2f*)&As[( 0 + l) * ALD + kk];
    v2f a1 = *(const v2f*)&As[(16 + l) * ALD + kk];
    v2f a2 = *(const v2f*)&As[(32 + l) * ALD + kk];
    acc0 = __builtin_amdgcn_wmma_f32_16x16x4_f32(false, a0, false, b, (short)0, acc0, false, false);
    acc1 = __builtin_amdgcn_wmma_f32_16x16x4_f32(false, a1, false, b, (short)0, acc1, false, false);
    acc2 = __builtin_amdgcn_wmma_f32_16x16x4_f32(false, a2, false, b, (short)0, acc2, false, false);
  }

  const int col = n0 + l;
  const float bcol = bias[col];
  v8f accs[3] = {acc0, acc1, acc2};
  #pragma unroll
  for (int mi = 0; mi < 3; ++mi) {
    #pragma unroll
    for (int i = 0; i < 8; ++i) {
      int row = m_base + mi * 16 + hf * 8 + i;
      if (row < M) {
        float v = scale * accs[mi][i] + bcol;
        if (res) v += res[(size_t)row * ldres + col];
        C[(size_t)row * ldc + col] = v;
      }
    }
  }
}

// ---------------------------------------------------------------------------
// Per-edge attention scores: wave per edge; lane d of head j via cross-lane
// reduction. ex = exp(score*pri/sqrt32); atomic denominator per (dst, head).
// (Max subtraction is omitted: softmax is shift-invariant and scores are O(1).)
// ---------------------------------------------------------------------------
__global__ __launch_bounds__(256) void edge_score(
    const float* __restrict__ P, const int* __restrict__ src, const int* __restrict__ dst,
    int kr_off, const float* __restrict__ rel_pri, int r,
    float* __restrict__ EX, float* __restrict__ DEN, int nE) {
  int wid = (int)(((size_t)blockIdx.x * blockDim.x + threadIdx.x) >> 5);
  int lane = threadIdx.x & 31;
  if (wid >= nE) return;
  int s = src[wid], d = dst[wid];
  const float* qrow = P + (size_t)d * PLD;
  const float* krow = P + (size_t)s * PLD + kr_off;
  float mine = 0.f;
  #pragma unroll
  for (int j = 0; j < 8; ++j) {
    float p = qrow[j * 32 + lane] * krow[j * 32 + lane];
    #pragma unroll
    for (int off = 16; off > 0; off >>= 1) p += __shfl_xor(p, off, 32);
    if (lane == j) mine = p;          // lane j keeps head-j score
  }
  if (lane < 8) {
    float sc = mine * rel_pri[r * 8 + lane] * 0.17677669529663687f; // 1/sqrt(32)
    float ex = expf(sc);
    EX[(size_t)wid * 8 + lane] = ex;
    atomicAdd(&DEN[(size_t)d * 8 + lane], ex);
  }
}

// ---------------------------------------------------------------------------
// Per-edge aggregation: attn = ex/den[dst]; t[dst] += vr[src] * attn (atomic).
// ---------------------------------------------------------------------------
__global__ __launch_bounds__(256) void edge_agg(
    const float* __restrict__ P, const int* __restrict__ src, const int* __restrict__ dst,
    int vr_off, const float* __restrict__ EX, const float* __restrict__ DEN,
    float* __restrict__ T, int nE) {
  int wid = (int)(((size_t)blockIdx.x * blockDim.x + threadIdx.x) >> 5);
  int lane = threadIdx.x & 31;
  if (wid >= nE) return;
  int s = src[wid], d = dst[wid];
  float a = 0.f;
  if (lane < 8) a = EX[(size_t)wid * 8 + lane] / DEN[(size_t)d * 8 + lane];
  const float* vrow = P + (size_t)s * PLD + vr_off;
  float* trow = T + (size_t)d * 256;
  #pragma unroll
  for (int j = 0; j < 8; ++j) {
    float aj = __shfl(a, j, 32);
    atomicAdd(&trow[j * 32 + lane], vrow[j * 32 + lane] * aj);
  }
}

// ---------------------------------------------------------------------------
// Row LayerNorm: wave per row (256 elems, 8 per lane), shfl reductions.
// ---------------------------------------------------------------------------
__global__ __launch_bounds__(256) void layernorm_rows(
    const float* __restrict__ Y, const float* __restrict__ g, const float* __restrict__ b,
    float* __restrict__ out, int M) {
  int row = (int)(((size_t)blockIdx.x * blockDim.x + threadIdx.x) >> 5);
  int lane = threadIdx.x & 31;
  if (row >= M) return;
  const float* y = Y + (size_t)row * 256;
  float x[8];
  float s = 0.f;
  #pragma unroll
  for (int j = 0; j < 8; ++j) { x[j] = y[j * 32 + lane]; s += x[j]; }
  #pragma unroll
  for (int off = 16; off > 0; off >>= 1) s += __shfl_xor(s, off, 32);
  float mean = s * (1.f / 256.f);
  float v = 0.f;
  #pragma unroll
  for (int j = 0; j < 8; ++j) { float dd = x[j] - mean; v += dd * dd; }
  #pragma unroll
  for (int off = 16; off > 0; off >>= 1) v += __shfl_xor(v, off, 32);
  float rstd = rsqrtf(v * (1.f / 256.f) + 1e-5f);
  #pragma unroll
  for (int j = 0; j < 8; ++j) {
    int c = j * 32 + lane;
    out[(size_t)row * 256 + c] = (x[j] - mean) * rstd * g[c] + b[c];
  }
}

extern "C" void kernel_launch(void* const* d_in, const int* in_sizes, int n_in,
                              void* d_out, int out_size, void* d_ws, size_t ws_size,
                              hipStream_t stream) {
  const float* h       = (const float*)d_in[0];
  const int*   src0    = (const int*)d_in[1];
  const int*   dst0    = (const int*)d_in[2];
  const int*   src1    = (const int*)d_in[3];
  const int*   dst1    = (const int*)d_in[4];
  const float* Wk      = (const float*)d_in[5];
  const float* bk      = (const float*)d_in[6];
  const float* Wq      = (const float*)d_in[7];
  const float* bq      = (const float*)d_in[8];
  const float* Wv      = (const float*)d_in[9];
  const float* bv      = (const float*)d_in[10];
  const float* Wa      = (const float*)d_in[11];
  const float* ba      = (const float*)d_in[12];
  const float* ln_g    = (const float*)d_in[13];
  const float* ln_b    = (const float*)d_in[14];
  const float* rel_pri = (const float*)d_in[15];
  const float* rel_att = (const float*)d_in[16];
  const float* rel_msg = (const float*)d_in[17];
  float* out = (float*)d_out;

  // workspace layout (floats); DEN0,DEN1,T contiguous for a single zero pass
  float* ws = (float*)d_ws;
  size_t o = 0;
  float* Wbig    = ws + o; o += (size_t)256 * 1280;
  float* BiasBig = ws + o; o += 1280;
  float* P       = ws + o; o += (size_t)N_NODES * 1280;
  float* EX0     = ws + o; o += (size_t)E_EDGES * 8;
  float* EX1     = ws + o; o += (size_t)E_EDGES * 8;
  float* DEN0    = ws + o; o += (size_t)N_NODES * 8;
  float* DEN1    = ws + o; o += (size_t)N_NODES * 8;
  float* T       = ws + o; o += (size_t)N_NODES * 256;
  float* Y       = ws + o; o += (size_t)N_NODES * 256;

  // 0) fused weights + biases
  build_wbig<<<(257 * 1280 + 255) / 256, 256, 0, stream>>>(
      Wk, bk, Wq, bq, Wv, bv, rel_att, rel_msg, Wbig, BiasBig);

  // zero DEN0 | DEN1 | T (contiguous region)
  zero_f32<<<2048, 256, 0, stream>>>(DEN0, (size_t)N_NODES * (8 + 8 + 256));

  // 1) fused projection GEMM: P[N,1280] = h @ Wbig + BiasBig
  dim3 g1((N_NODES + MB - 1) / MB, 1280 / 128);
  wmma_gemm_k256<<<g1, 256, 0, stream>>>(h, 256, Wbig, 1280, BiasBig,
                                         nullptr, 0, 1.0f, P, 1280, N_NODES);

  // 2) edge attention + aggregation (relation 0 and 1); offsets into P columns
  int eblocks = (E_EDGES + 7) / 8;  // 8 waves per 256-thread block
  edge_score<<<eblocks, 256, 0, stream>>>(P, src0, dst0, 256, rel_pri, 0, EX0, DEN0, E_EDGES);
  edge_score<<<eblocks, 256, 0, stream>>>(P, src1, dst1, 512, rel_pri, 1, EX1, DEN1, E_EDGES);
  edge_agg  <<<eblocks, 256, 0, stream>>>(P, src0, dst0, 768,  EX0, DEN0, T, E_EDGES);
  edge_agg  <<<eblocks, 256, 0, stream>>>(P, src1, dst1, 1024, EX1, DEN1, T, E_EDGES);

  // 3) output GEMM: Y = 0.5*(T @ Wa) + ba + h   (cross-relation mean folded into scale)
  dim3 g3((N_NODES + MB - 1) / MB, 256 / 128);
  wmma_gemm_k256<<<g3, 256, 0, stream>>>(T, 256, Wa, 256, ba,
                                         h, 256, 0.5f, Y, 256, N_NODES);

  // 4) LayerNorm -> out
  layernorm_rows<<<(N_NODES + 7) / 8, 256, 0, stream>>>(Y, ln_g, ln_b, out, N_NODES);
}